// DTDFII_37641093382425
// MI455X (gfx1250) — compile-verified
//
#include <hip/hip_runtime.h>

typedef __attribute__((ext_vector_type(2))) float v2f;
typedef __attribute__((ext_vector_type(4))) float v4f;
typedef __attribute__((ext_vector_type(8))) float v8f;

// Problem constants (reference: B=64, T=262144, x[B][T], out[B][T])
constexpr int Bc     = 64;
constexpr int Tc     = 262144;
constexpr int CHUNK  = 1024;          // time steps per scan chunk
constexpr int NCHUNK = Tc / CHUNK;    // 256 chunks per channel
constexpr int NBLK   = CHUNK / 16;    // 64 16-step blocks per chunk

// Workspace layout (float offsets)
constexpr int OFF_H   = 0;                         // h[16] impulse response
constexpr int OFF_G   = 16;                        // g[t] = e0^T A^t, 16x2
constexpr int OFF_W   = 48;                        // W[2][16], W[:,j] = A^(15-j) c
constexpr int OFF_M16 = 80;                        // A^16 (2x2 row-major)
constexpr int OFF_MC  = 84;                        // A^CHUNK (2x2 row-major)
constexpr int OFF_WQ0 = 96;                        // wq0[i] = (A^(CHUNK-1-i) c).x
constexpr int OFF_WQ1 = OFF_WQ0 + CHUNK;           // wq1[i] = (A^(CHUNK-1-i) c).y
constexpr int OFF_D   = OFF_WQ1 + CHUNK;           // d[NCHUNK][Bc][2]
constexpr int OFF_S   = OFF_D + NCHUNK * Bc * 2;   // chunk-start states [NCHUNK][Bc][2]

// ---------------- Phase A: derive all coefficient tables (1 thread) -----------
__global__ void dtdfii_phaseA(const float* __restrict__ b0p, const float* __restrict__ b1p,
                              const float* __restrict__ b2p, const float* __restrict__ a1p,
                              const float* __restrict__ a2p, float* __restrict__ ws) {
    const float b0 = b0p[0], b1 = b1p[0], b2 = b2p[0], a1 = a1p[0], a2 = a2p[0];
    const float c0 = b1 - a1 * b0;
    const float c1 = b2 - a2 * b0;

    // h[0]=b0, h[k]=e0^T A^(k-1) c ; g[t]=e0^T A^t  (p iterates p<-pA)
    ws[OFF_H + 0] = b0;
    float p0 = 1.f, p1 = 0.f;
    for (int t = 0; t < 16; ++t) {
        ws[OFF_G + 2 * t + 0] = p0;
        ws[OFF_G + 2 * t + 1] = p1;
        if (t < 15) ws[OFF_H + t + 1] = fmaf(p0, c0, p1 * c1);
        const float np0 = fmaf(-a1, p0, -a2 * p1);
        const float np1 = p0;
        p0 = np0; p1 = np1;
    }

    // q[k] = A^k c ; wq (reversed) for chunk dot; W[:,j] = q[15-j]
    float q0 = c0, q1 = c1;
    for (int k = 0; k < CHUNK; ++k) {
        const int i = CHUNK - 1 - k;
        ws[OFF_WQ0 + i] = q0;
        ws[OFF_WQ1 + i] = q1;
        if (k < 16) {
            ws[OFF_W + (15 - k)]      = q0;
            ws[OFF_W + 16 + (15 - k)] = q1;
        }
        const float nq0 = fmaf(-a1, q0, q1);
        const float nq1 = -a2 * q0;
        q0 = nq0; q1 = nq1;
    }

    // Matrix powers: A^16 and A^CHUNK
    float m00 = 1.f, m01 = 0.f, m10 = 0.f, m11 = 1.f;
    for (int k = 0; k < CHUNK; ++k) {
        const float n00 = fmaf(-a1, m00, m10);
        const float n01 = fmaf(-a1, m01, m11);
        const float n10 = -a2 * m00;
        const float n11 = -a2 * m01;
        m00 = n00; m01 = n01; m10 = n10; m11 = n11;
        if (k == 15) {
            ws[OFF_M16 + 0] = m00; ws[OFF_M16 + 1] = m01;
            ws[OFF_M16 + 2] = m10; ws[OFF_M16 + 3] = m11;
        }
    }
    ws[OFF_MC + 0] = m00; ws[OFF_MC + 1] = m01;
    ws[OFF_MC + 2] = m10; ws[OFF_MC + 3] = m11;
}

// ---------------- Phase 1: per-(channel,chunk) d = two weighted dots ----------
__global__ void dtdfii_phase1(const float* __restrict__ x, float* __restrict__ ws) {
    const int tid  = blockIdx.x * blockDim.x + threadIdx.x;
    const int wave = tid >> 5;
    const int lane = tid & 31;
    const int ch = wave / NCHUNK;
    const int k  = wave % NCHUNK;

    const float* __restrict__ wq0 = ws + OFF_WQ0;
    const float* __restrict__ wq1 = ws + OFF_WQ1;
    const float* __restrict__ xp  = x + (size_t)ch * Tc + (size_t)k * CHUNK;

    float s0 = 0.f, s1 = 0.f;
    #pragma unroll 4
    for (int i = lane; i < CHUNK; i += 32) {
        const float xv = xp[i];
        s0 = fmaf(wq0[i], xv, s0);
        s1 = fmaf(wq1[i], xv, s1);
    }
    #pragma unroll
    for (int m = 16; m >= 1; m >>= 1) {
        s0 += __shfl_xor(s0, m, 32);
        s1 += __shfl_xor(s1, m, 32);
    }
    if (lane == 0) {
        ws[OFF_D + (k * Bc + ch) * 2 + 0] = s0;
        ws[OFF_D + (k * Bc + ch) * 2 + 1] = s1;
    }
}

// ---------------- Phase 2: affine scan over chunks (per channel) --------------
__global__ void dtdfii_phase2(float* __restrict__ ws) {
    const int ch = threadIdx.x;
    if (ch >= Bc) return;
    const float m00 = ws[OFF_MC + 0], m01 = ws[OFF_MC + 1];
    const float m10 = ws[OFF_MC + 2], m11 = ws[OFF_MC + 3];
    float s0 = 0.f, s1 = 0.f;
    for (int k = 0; k < NCHUNK; ++k) {
        ws[OFF_S + (k * Bc + ch) * 2 + 0] = s0;
        ws[OFF_S + (k * Bc + ch) * 2 + 1] = s1;
        const float d0 = ws[OFF_D + (k * Bc + ch) * 2 + 0];
        const float d1 = ws[OFF_D + (k * Bc + ch) * 2 + 1];
        const float t0 = fmaf(m00, s0, fmaf(m01, s1, d0));
        const float t1 = fmaf(m10, s0, fmaf(m11, s1, d1));
        s0 = t0; s1 = t1;
    }
}

// ---------------- Phase 3: WMMA replay. One wave = 16 channels x 1 chunk ------
// Per 16-time block:  Out(16t x 16ch) = T*X + G*S   via 5x v_wmma_f32_16x16x4_f32
// State hop:          S <- M16*S + W*X              via VALU + shfl_xor(16)
__global__ void dtdfii_phase3(const float* __restrict__ x, float* __restrict__ out,
                              const float* __restrict__ ws) {
    const int tid  = blockIdx.x * blockDim.x + threadIdx.x;
    const int wave = tid >> 5;            // 0 .. 4*NCHUNK-1
    const int lane = tid & 31;
    const int cg   = wave / NCHUNK;       // channel group (4 groups of 16)
    const int k    = wave % NCHUNK;       // chunk index
    const int half = lane >> 4;           // 0: K/M-cols 0-1 & rows 0-7 ; 1: 2-3 & 8-15
    const int n    = lane & 15;           // row (A) / column-channel (B,C,D)
    const int ch   = cg * 16 + n;

    // --- per-lane constant panels -------------------------------------------
    // A-panels of Toeplitz T: tA[p] = (T[n][4p+2h], T[n][4p+2h+1]), T[t][j]=h[t-j]
    v2f tA[4];
    #pragma unroll
    for (int p = 0; p < 4; ++p) {
        const int j0 = 4 * p + 2 * half;
        tA[p].x = (j0     <= n) ? ws[OFF_H + (n - j0)]     : 0.f;
        tA[p].y = (j0 + 1 <= n) ? ws[OFF_H + (n - j0 - 1)] : 0.f;
    }
    // A-panel of G (16x2 padded to 16x4): cols 2,3 are zero
    v2f gA;
    gA.x = half ? 0.f : ws[OFF_G + 2 * n + 0];
    gA.y = half ? 0.f : ws[OFF_G + 2 * n + 1];
    // W coefficients this lane needs for the state dot (its 8 time slots)
    v2f w0[4], w1[4];
    #pragma unroll
    for (int p = 0; p < 4; ++p) {
        const int j0 = 4 * p + 2 * half;
        w0[p].x = ws[OFF_W + j0];          w0[p].y = ws[OFF_W + j0 + 1];
        w1[p].x = ws[OFF_W + 16 + j0];     w1[p].y = ws[OFF_W + 16 + j0 + 1];
    }
    const float m00 = ws[OFF_M16 + 0], m01 = ws[OFF_M16 + 1];
    const float m10 = ws[OFF_M16 + 2], m11 = ws[OFF_M16 + 3];

    // chunk-start state for this channel (held by all 32 lanes)
    float s0 = ws[OFF_S + (k * Bc + ch) * 2 + 0];
    float s1 = ws[OFF_S + (k * Bc + ch) * 2 + 1];

    const size_t chBase = (size_t)ch * Tc;
    const size_t tBase  = (size_t)k * CHUNK;

    for (int b = 0; b < NBLK; ++b) {
        const size_t t0 = tBase + (size_t)b * 16;

        // B-panels of X (time x channel): xB[p] = (X[4p+2h][n], X[4p+2h+1][n])
        v2f xB[4];
        #pragma unroll
        for (int p = 0; p < 4; ++p)
            xB[p] = *reinterpret_cast<const v2f*>(x + chBase + t0 + 4 * p + 2 * half);

        // B-panel of S (2x16 padded to 4x16): rows 2,3 zero
        v2f sB;
        sB.x = half ? 0.f : s0;
        sB.y = half ? 0.f : s1;

        v8f acc = {};
        acc = __builtin_amdgcn_wmma_f32_16x16x4_f32(false, gA,    false, sB,    (short)0, acc, false, false);
        #pragma unroll
        for (int p = 0; p < 4; ++p)
            acc = __builtin_amdgcn_wmma_f32_16x16x4_f32(false, tA[p], false, xB[p], (short)0, acc, false, false);

        // store Out: lane holds rows (half?8:0)+0..7 (= times), column = channel
        float* op = out + chBase + t0 + 8 * half;
        v4f lo = {acc[0], acc[1], acc[2], acc[3]};
        v4f hi = {acc[4], acc[5], acc[6], acc[7]};
        *reinterpret_cast<v4f*>(op)     = lo;
        *reinterpret_cast<v4f*>(op + 4) = hi;

        // state hop: S <- M16*S + W*X  (dot split across lane n / n+16)
        float d0 = 0.f, d1 = 0.f;
        #pragma unroll
        for (int p = 0; p < 4; ++p) {
            d0 = fmaf(w0[p].x, xB[p].x, fmaf(w0[p].y, xB[p].y, d0));
            d1 = fmaf(w1[p].x, xB[p].x, fmaf(w1[p].y, xB[p].y, d1));
        }
        d0 += __shfl_xor(d0, 16, 32);
        d1 += __shfl_xor(d1, 16, 32);
        const float ns0 = fmaf(m00, s0, fmaf(m01, s1, d0));
        const float ns1 = fmaf(m10, s0, fmaf(m11, s1, d1));
        s0 = ns0; s1 = ns1;
    }
}

extern "C" void kernel_launch(void* const* d_in, const int* in_sizes, int n_in,
                              void* d_out, int out_size, void* d_ws, size_t ws_size,
                              hipStream_t stream) {
    const float* x  = (const float*)d_in[0];
    const float* b0 = (const float*)d_in[1];
    const float* b1 = (const float*)d_in[2];
    const float* b2 = (const float*)d_in[3];
    const float* a1 = (const float*)d_in[4];
    const float* a2 = (const float*)d_in[5];
    float* out = (float*)d_out;
    float* ws  = (float*)d_ws;

    dtdfii_phaseA<<<1, 1, 0, stream>>>(b0, b1, b2, a1, a2, ws);
    dtdfii_phase1<<<(Bc * NCHUNK) / 8, 256, 0, stream>>>(x, ws);   // 8 waves/block
    dtdfii_phase2<<<1, 64, 0, stream>>>(ws);
    dtdfii_phase3<<<(4 * NCHUNK) / 8, 256, 0, stream>>>(x, out, ws);
}